// YOLOLoss_21345987461723
// MI455X (gfx1250) — compile-verified
//
#include <hip/hip_runtime.h>
#include <math.h>

// ---------------------------------------------------------------------------
// YOLO loss for MI455X (gfx1250, wave32).
// d_out = [ pred passthrough : 47,132,160 f32 ][ total_loss : 1 f32 ]
// Dominant cost: the 377 MB passthrough (~16 us @ 23.3 TB/s). Loss side is a
// single 2.2 MB conf-plane reduction + ~87 KB of scattered obj-cell reads.
// ---------------------------------------------------------------------------

namespace {
constexpr int  kNB = 32, kNA = 3, kNC = 80, kNG = 76, kNT = 256;
constexpr int  kGG       = kNG * kNG;                    // 5776
constexpr int  kSpatial  = kNA * kGG;                    // 17328
constexpr long kPredPerB = (long)(kNC + 5) * kSpatial;   // 1,472,880
constexpr long kTotalCells = (long)kNB * kSpatial;       // 554,496
constexpr long kPredElems  = (long)kNB * kPredPerB;      // 47,132,160
constexpr float kStride = 8.0f;                          // 608 / 76
constexpr float kIgnore = 0.5f;
}

typedef float v2f __attribute__((ext_vector_type(2)));
typedef float v4f __attribute__((ext_vector_type(4)));
typedef float v8f __attribute__((ext_vector_type(8)));

__device__ __forceinline__ float clip_p(float p) {
  return fminf(fmaxf(p, 1e-12f), 1.0f - 1e-12f);
}

// Wave32 sum. First halving stage done on the matrix pipe:
// A = ones(16x4), B = {x, 0} -> D[m][n] = x_n + x_{n+16} (exact f32 FMAs),
// broadcast across all lanes/rows of the C/D layout, so c[0] in lane L holds
// x_{L%16} + x_{L%16+16}. Remaining 16-lane tree via shfl_xor.
__device__ __forceinline__ float wave_sum32(float x) {
#if __has_builtin(__builtin_amdgcn_wmma_f32_16x16x4_f32)
  v2f a; a.x = 1.0f; a.y = 1.0f;   // A-matrix: all ones (16x4)
  v2f b; b.x = x;    b.y = 0.0f;   // B-matrix rows K0/K1 = lane values, K2/K3 = 0
  v8f c = {};
  c = __builtin_amdgcn_wmma_f32_16x16x4_f32(false, a, false, b,
                                            (short)0, c, false, false);
  float s = c[0];
#else
  float s = x + __shfl_xor(x, 16, 32);
#endif
  s += __shfl_xor(s, 8, 32);
  s += __shfl_xor(s, 4, 32);
  s += __shfl_xor(s, 2, 32);
  s += __shfl_xor(s, 1, 32);
  return s;
}

// ---------------------------------------------------------------------------
// Pass-through copy: 128-bit non-temporal streaming (write-once data, keep L2
// clean). Grid-stride, coalesced.
// ---------------------------------------------------------------------------
__global__ void k_copy(const v4f* __restrict__ src, v4f* __restrict__ dst, long n4) {
  long i    = (long)blockIdx.x * blockDim.x + threadIdx.x;
  long step = (long)gridDim.x * blockDim.x;
  for (; i < n4; i += step) {
    v4f v = __builtin_nontemporal_load(src + i);
    __builtin_nontemporal_store(v, dst + i);
  }
}

__global__ void k_init(float* acc) {
  if (threadIdx.x < 16) acc[threadIdx.x] = 0.0f;
}

// ---------------------------------------------------------------------------
// Sum of -log(1-conf) over ALL cells (noobj BCE before removing excluded
// cells). Conf plane is channel 4: pred[b*kPredPerB + 4*kSpatial + r].
// acc[0] += block partials.
// ---------------------------------------------------------------------------
__global__ void k_conf(const float* __restrict__ pred, float* __restrict__ acc) {
  constexpr int E = 8;
  long base = ((long)blockIdx.x * blockDim.x) * E + threadIdx.x;
  float s = 0.0f;
  for (int e = 0; e < E; ++e) {
    long i = base + (long)e * blockDim.x;
    if (i < kTotalCells) {
      long b = i / kSpatial;
      long r = i - b * kSpatial;
      float p = clip_p(pred[b * kPredPerB + 4l * kSpatial + r]);
      s += -logf(1.0f - p);
    }
  }
  __shared__ float wsum[8];
  float w = wave_sum32(s);                 // EXEC all-ones: uniform call
  int wid = threadIdx.x >> 5, lane = threadIdx.x & 31;
  if (lane == 0) wsum[wid] = w;
  __syncthreads();
  if (wid == 0) {
    float t = (lane < (int)(blockDim.x >> 5)) ? wsum[lane] : 0.0f;
    t = wave_sum32(t);
    if (lane == 0) atomicAdd(&acc[0], t);
  }
}

// ---------------------------------------------------------------------------
// Target-side logic, one block of NT=256 threads (one thread per target).
//   acc[1] = #distinct noobj-excluded cells    acc[2] = their -log(1-conf) sum
//   acc[3] = #distinct obj cells               acc[4] = obj -log(conf) sum
//   acc[5..8] = sum (px-tx)^2, (py-ty)^2, (pw-tw)^2, (ph-th)^2 over obj cells
//   acc[9] = class-BCE sum over obj cells x 80 classes
// ---------------------------------------------------------------------------
__global__ void k_targets(const float* __restrict__ pred,
                          const float* __restrict__ targ,
                          float* __restrict__ acc) {
  __shared__ int s_cell[kNT];
  __shared__ int s_lab[kNT];
  __shared__ int s_excl[kNT * kNA];

  const float AW[3]   = {1.25f, 2.0f, 4.125f};   // anchors / stride
  const float AH[3]   = {1.625f, 3.75f, 2.875f};
  const float AWpx[3] = {10.f, 16.f, 33.f};
  const float AHpx[3] = {13.f, 30.f, 23.f};

  int t = threadIdx.x;
  const float* tr = targ + t * 6;
  int   b   = (int)tr[0];
  int   lab = (int)tr[1];
  float gx = tr[2] * kNG, gy = tr[3] * kNG;
  float gw = tr[4] * kNG, gh = tr[5] * kNG;

  float iou[3]; float best = -1.0f; int bn = 0;
  for (int a = 0; a < 3; ++a) {
    float inter = fminf(AW[a], gw) * fminf(AH[a], gh);
    float uni   = AW[a] * AH[a] + gw * gh - inter + 1e-16f;
    iou[a] = inter / uni;
    if (iou[a] > best) { best = iou[a]; bn = a; }   // first-max = jnp.argmax
  }
  int gi = min(max((int)gx, 0), kNG - 1);
  int gj = min(max((int)gy, 0), kNG - 1);
  int cell = ((b * kNA + bn) * kNG + gj) * kNG + gi;
  s_cell[t] = cell;
  s_lab[t]  = lab;
  for (int a = 0; a < 3; ++a)
    s_excl[t * 3 + a] = (a == bn || iou[a] > kIgnore)
        ? ((b * kNA + a) * kNG + gj) * kNG + gi : -1;
  __syncthreads();

  // Distinct noobj-excluded cells (first occurrence owns the contribution).
  float exCnt = 0.0f, exSum = 0.0f;
  for (int a = 0; a < 3; ++a) {
    int e = t * 3 + a;
    int c = s_excl[e];
    if (c < 0) continue;
    bool first = true;
    for (int u = 0; u < e; ++u) if (s_excl[u] == c) { first = false; break; }
    if (first) {
      int  bb = c / kSpatial;
      int  r  = c - bb * kSpatial;
      float p = clip_p(pred[(long)bb * kPredPerB + 4l * kSpatial + r]);
      exSum += -logf(1.0f - p);
      exCnt += 1.0f;
    }
  }
  if (exCnt != 0.0f) { atomicAdd(&acc[1], exCnt); atomicAdd(&acc[2], exSum); }

  // Representative target per obj cell: last scatter wins (.set semantics).
  bool rep = true;
  for (int u = t + 1; u < kNT; ++u) if (s_cell[u] == cell) { rep = false; break; }
  if (rep) {
    long base = (long)b * kPredPerB;
    long off  = (long)bn * kGG + (long)gj * kNG + gi;

    float pc = clip_p(pred[base + 4l * kSpatial + off]);
    atomicAdd(&acc[3], 1.0f);
    atomicAdd(&acc[4], -logf(pc));

    float p0 = pred[base + 0l * kSpatial + off];
    float p1 = pred[base + 1l * kSpatial + off];
    float p2 = pred[base + 2l * kSpatial + off];
    float p3 = pred[base + 3l * kSpatial + off];
    float px = (p0 / kStride - (float)gi + 0.5f) * 0.5f;
    float py = (p1 / kStride - (float)gj + 0.5f) * 0.5f;
    float pw = sqrtf(p2 / AWpx[bn]) * 0.5f;
    float ph = sqrtf(p3 / AHpx[bn]) * 0.5f;
    float tx = gx - floorf(gx);
    float ty = (gy - floorf(gy) + 0.5f) * 0.5f;
    float tw = sqrtf(gw / AW[bn]) * 0.5f;
    float th = sqrtf(gh / AH[bn]) * 0.5f;
    atomicAdd(&acc[5], (px - tx) * (px - tx));
    atomicAdd(&acc[6], (py - ty) * (py - ty));
    atomicAdd(&acc[7], (pw - tw) * (pw - tw));
    atomicAdd(&acc[8], (ph - th) * (ph - th));

    // tcls: every duplicate target's label sets a 1 at this cell.
    unsigned m0 = 0, m1 = 0, m2 = 0;
    for (int u = 0; u < kNT; ++u) {
      if (s_cell[u] == cell) {
        int l = s_lab[u];
        if (l < 32)      m0 |= 1u << l;
        else if (l < 64) m1 |= 1u << (l - 32);
        else             m2 |= 1u << (l - 64);
      }
    }
    float cs = 0.0f;
    for (int c = 0; c < kNC; ++c) {
      float p = clip_p(pred[base + (long)(5 + c) * kSpatial + off]);
      unsigned bit = (c < 32) ? (m0 >> c)
                   : (c < 64) ? (m1 >> (c - 32))
                              : (m2 >> (c - 64));
      float tt = (float)(bit & 1u);
      cs += -(tt * logf(p) + (1.0f - tt) * logf(1.0f - p));
    }
    atomicAdd(&acc[9], cs);
  }
}

__global__ void k_final(const float* __restrict__ acc, float* __restrict__ out) {
  float objc = acc[3];
  float d1   = fmaxf(objc, 1.0f);
  float noobj_sum = acc[0] - acc[2];                           // all minus excluded
  float noobj_cnt = fmaxf((float)kTotalCells - acc[1], 1.0f);
  float lx = acc[5] / d1, ly = acc[6] / d1;
  float lw = acc[7] / d1, lh = acc[8] / d1;
  float lconf = acc[4] / d1 + 100.0f * (noobj_sum / noobj_cnt);
  float lcls  = acc[9] / fmaxf(objc * (float)kNC, 1.0f);
  out[kPredElems] = (lx + ly) * 10.0f + (lw + lh) * 10.0f + lconf * 1.0f + lcls * 10.0f;
}

extern "C" void kernel_launch(void* const* d_in, const int* in_sizes, int n_in,
                              void* d_out, int out_size, void* d_ws, size_t ws_size,
                              hipStream_t stream) {
  (void)in_sizes; (void)n_in; (void)out_size; (void)ws_size;
  const float* pred = (const float*)d_in[0];
  const float* targ = (const float*)d_in[1];
  float* out = (float*)d_out;
  float* acc = (float*)d_ws;   // 16 f32 accumulators

  k_init<<<1, 32, 0, stream>>>(acc);

  {
    int blocks = (int)((kTotalCells + 256l * 8 - 1) / (256l * 8));   // 271
    k_conf<<<blocks, 256, 0, stream>>>(pred, acc);
  }
  k_targets<<<1, kNT, 0, stream>>>(pred, targ, acc);
  k_final<<<1, 1, 0, stream>>>(acc, out);

  {
    long n4 = kPredElems / 4;                                        // 11,783,040
    int blocks = (int)((n4 + 256l * 8 - 1) / (256l * 8));            // ~5754
    k_copy<<<blocks, 256, 0, stream>>>((const v4f*)pred, (v4f*)out, n4);
  }
}